// Linear_50568944943395
// MI455X (gfx1250) — compile-verified
//
#include <hip/hip_runtime.h>

// ---------------------------------------------------------------------------
// Fused sparse-embedding + varlen-pooling + dense-linear kernel for gfx1250.
//
// Data-movement policy (MI455X: 23.3 TB/s HBM, 192 MB L2):
//   * embedding tables (80 MB total) fit in L2 -> gather with default RT hint
//   * index streams (60 MB) + dense vals + output -> non-temporal (TH_NT), so
//     the streamed traffic does not evict the L2-resident tables.
//   * dense GEMV (K=13, N=1) done with V_WMMA_F32_16X16X4_F32 per 16-row tile;
//     K padded to 16 by zeroing B rows 13..15 (branch-free, no exec masking).
// ---------------------------------------------------------------------------

typedef int   v2i __attribute__((ext_vector_type(2)));
typedef int   v4i __attribute__((ext_vector_type(4)));
typedef float v2f __attribute__((ext_vector_type(2)));
typedef float v8f __attribute__((ext_vector_type(8)));

#define N_SPARSE 16
#define N_VARLEN 4
#define MAXLEN   50
#define N_DENSE  13
#define VOCAB_SZ 1000000

__global__ __launch_bounds__(256)
void fused_rec_linear_kernel(const int* __restrict__ sparse_ids,   // [B,16]
                             const int* __restrict__ varlen_ids,   // [B,4,50]
                             const float* __restrict__ dense_vals, // [B,13]
                             const float* __restrict__ emb_sparse, // [16,VOCAB]
                             const float* __restrict__ emb_varlen, // [4,VOCAB]
                             const float* __restrict__ dense_w,    // [13,1]
                             float* __restrict__ out)              // [B,1]
{
    __shared__ float dense_stage[256];

    const int tid  = threadIdx.x;
    const int lane = tid & 31;
    const int b    = blockIdx.x * 256 + tid;

    // =====================================================================
    // Dense GEMV via WMMA: wave handles 32 rows = two 16x16x4-chain tiles.
    // A (16x4 f32): lane L feeds row M=L&15; lanes 16..31 feed K+2,K+3.
    // B (4x16 f32): B[k][n] = w[k] broadcast over n -> every column of D is
    // the per-row dot product.
    // =====================================================================
    const int waveBase = blockIdx.x * 256 + (tid & ~31);
    const int m        = lane & 15;
    const int kAdd     = (lane >> 4) << 1;     // 0 for lanes 0-15, 2 for 16-31

    v8f c0 = {0.f, 0.f, 0.f, 0.f, 0.f, 0.f, 0.f, 0.f};
    v8f c1 = {0.f, 0.f, 0.f, 0.f, 0.f, 0.f, 0.f, 0.f};

    const float* r0 = dense_vals + (size_t)(waveBase + m) * N_DENSE;
    const float* r1 = r0 + (size_t)16 * N_DENSE;

    // k = 0..11: every lane's K indices are in range -> fully unconditional.
#pragma unroll
    for (int kb = 0; kb < 12; kb += 4) {
        const int k0 = kb + kAdd;

        v2f bm;
        bm.x = dense_w[k0];
        bm.y = dense_w[k0 + 1];

        v2f a0, a1;
        a0.x = __builtin_nontemporal_load(r0 + k0);
        a0.y = __builtin_nontemporal_load(r0 + k0 + 1);
        a1.x = __builtin_nontemporal_load(r1 + k0);
        a1.y = __builtin_nontemporal_load(r1 + k0 + 1);

        c0 = __builtin_amdgcn_wmma_f32_16x16x4_f32(false, a0, false, bm,
                                                   (short)0, c0, false, false);
        c1 = __builtin_amdgcn_wmma_f32_16x16x4_f32(false, a1, false, bm,
                                                   (short)0, c1, false, false);
    }

    // Tail k-block 12..15: only k=12 is real.  Zero the B rows for k=13,14,15
    // (a zero B row kills the product regardless of A), so the A loads can be
    // unconditional at the clamped in-bounds offset +12 -> no exec masking.
    {
        const float w12 = dense_w[12];
        v2f bm;
        bm.x = (kAdd == 0) ? w12 : 0.f;   // B row 12 lives on lanes 0-15
        bm.y = 0.f;                       // B rows 13 / 15 are zero

        v2f a0, a1;
        a0.x = __builtin_nontemporal_load(r0 + 12);  // lanes>=16: k=14 slot,
        a0.y = 0.f;                                  // nulled by zero B row
        a1.x = __builtin_nontemporal_load(r1 + 12);
        a1.y = 0.f;

        c0 = __builtin_amdgcn_wmma_f32_16x16x4_f32(false, a0, false, bm,
                                                   (short)0, c0, false, false);
        c1 = __builtin_amdgcn_wmma_f32_16x16x4_f32(false, a1, false, bm,
                                                   (short)0, c1, false, false);
    }

    // D columns identical: lanes 0-15 hold tile rows 0-7 in c[0..7],
    // lanes 16-31 hold rows 8-15.  Lanes 0 and 16 stage 32 results to LDS.
    const int wslot = tid & ~31;
    if (lane == 0) {
#pragma unroll
        for (int r = 0; r < 8; ++r) {
            dense_stage[wslot + r]      = c0[r];   // tile0 rows 0-7
            dense_stage[wslot + 16 + r] = c1[r];   // tile1 rows 0-7
        }
    }
    if (lane == 16) {
#pragma unroll
        for (int r = 0; r < 8; ++r) {
            dense_stage[wslot + 8 + r]  = c0[r];   // tile0 rows 8-15
            dense_stage[wslot + 24 + r] = c1[r];   // tile1 rows 8-15
        }
    }
    __syncthreads();

    // =====================================================================
    // Sparse 1-d embeddings: 16 independent gathers (tables resident in L2).
    // Index stream loaded non-temporally (b128).
    // =====================================================================
    const v4i* sid = (const v4i*)(sparse_ids + (size_t)b * N_SPARSE);
    v4i s0 = __builtin_nontemporal_load(sid + 0);
    v4i s1 = __builtin_nontemporal_load(sid + 1);
    v4i s2 = __builtin_nontemporal_load(sid + 2);
    v4i s3 = __builtin_nontemporal_load(sid + 3);

    float acc0, acc1, acc2, acc3;
    acc0  = emb_sparse[(size_t)0  * VOCAB_SZ + s0.x];
    acc1  = emb_sparse[(size_t)1  * VOCAB_SZ + s0.y];
    acc2  = emb_sparse[(size_t)2  * VOCAB_SZ + s0.z];
    acc3  = emb_sparse[(size_t)3  * VOCAB_SZ + s0.w];
    acc0 += emb_sparse[(size_t)4  * VOCAB_SZ + s1.x];
    acc1 += emb_sparse[(size_t)5  * VOCAB_SZ + s1.y];
    acc2 += emb_sparse[(size_t)6  * VOCAB_SZ + s1.z];
    acc3 += emb_sparse[(size_t)7  * VOCAB_SZ + s1.w];
    acc0 += emb_sparse[(size_t)8  * VOCAB_SZ + s2.x];
    acc1 += emb_sparse[(size_t)9  * VOCAB_SZ + s2.y];
    acc2 += emb_sparse[(size_t)10 * VOCAB_SZ + s2.z];
    acc3 += emb_sparse[(size_t)11 * VOCAB_SZ + s2.w];
    acc0 += emb_sparse[(size_t)12 * VOCAB_SZ + s3.x];
    acc1 += emb_sparse[(size_t)13 * VOCAB_SZ + s3.y];
    acc2 += emb_sparse[(size_t)14 * VOCAB_SZ + s3.z];
    acc3 += emb_sparse[(size_t)15 * VOCAB_SZ + s3.w];

    // =====================================================================
    // Varlen features: 4 tables x 50 ids, masked sum-pool (id 0 == pad).
    // Gather unconditionally (address 0 is valid table memory) and select by
    // the mask -> no divergence, loads stay batched for max MLP.
    // Row byte offset is 200-aligned -> use b64 index loads.
    // =====================================================================
#pragma unroll 1
    for (int v = 0; v < N_VARLEN; ++v) {
        const float* tab = emb_varlen + (size_t)v * VOCAB_SZ;
        const v2i* ids =
            (const v2i*)(varlen_ids + ((size_t)b * N_VARLEN + v) * MAXLEN);
        float pa = 0.f, pb = 0.f;
#pragma unroll
        for (int j = 0; j < MAXLEN / 2; ++j) {
            v2i p = __builtin_nontemporal_load(ids + j);
            float g0 = tab[p.x];
            float g1 = tab[p.y];
            pa += (p.x > 0) ? g0 : 0.f;
            pb += (p.y > 0) ? g1 : 0.f;
        }
        acc2 += pa;
        acc3 += pb;
    }

    const float total = (acc0 + acc1) + (acc2 + acc3) + dense_stage[tid];
    __builtin_nontemporal_store(total, out + b);
}

// ---------------------------------------------------------------------------
// Host launch
// ---------------------------------------------------------------------------
extern "C" void kernel_launch(void* const* d_in, const int* in_sizes, int n_in,
                              void* d_out, int out_size, void* d_ws, size_t ws_size,
                              hipStream_t stream)
{
    const int*   sparse_ids = (const int*)d_in[0];
    const int*   varlen_ids = (const int*)d_in[1];
    const float* dense_vals = (const float*)d_in[2];
    const float* emb_sparse = (const float*)d_in[3];
    const float* emb_varlen = (const float*)d_in[4];
    const float* dense_w    = (const float*)d_in[5];
    float*       out        = (float*)d_out;

    const int B = in_sizes[0] / N_SPARSE;   // 65536
    dim3 grid(B / 256), block(256);
    hipLaunchKernelGGL(fused_rec_linear_kernel, grid, block, 0, stream,
                       sparse_ids, varlen_ids, dense_vals,
                       emb_sparse, emb_varlen, dense_w, out);
}